// NonLocalMeans_75058848465573
// MI455X (gfx1250) — compile-verified
//
#include <hip/hip_runtime.h>
#include <math.h>

typedef __attribute__((ext_vector_type(2))) float v2f;
typedef __attribute__((ext_vector_type(4))) float v4f;
typedef __attribute__((ext_vector_type(8))) float v8f;

#define TILE      32
#define HALO      7                 // search radius 5 + patch radius 2
#define YDIM      (TILE + 2*HALO)   // 46
#define Y_STRIDE  47
#define ZDIM      (TILE + 4)        // 36 rowsum rows (tile + 2*patch radius)
#define RS_STRIDE 33
#define W_IMG     1024
#define MASKW     1023

__global__ __launch_bounds__(128)
void nlm_kernel(const float* __restrict__ rgb, float* __restrict__ out) {
    __shared__ v4f   sC[YDIM * Y_STRIDE];    // (r, g, b, 1.0)  34592 B
    __shared__ float sY[YDIM * Y_STRIDE];    // luminance        8648 B
    __shared__ float sRS[ZDIM * RS_STRIDE];  // horizontal 5-tap 4752 B

    const int tid  = threadIdx.x;
    const int lane = tid & 31;
    const int wave = tid >> 5;

    const int blk  = blockIdx.x;
    const int b    = blk >> 10;          // batch index (1024 tiles per image)
    const int t    = blk & 1023;
    const int row0 = (t >> 5) * TILE;
    const int col0 = (t & 31) * TILE;

    const size_t plane = (size_t)W_IMG * W_IMG;
    const float* pR = rgb + ((size_t)b * 3 + 0) * plane;
    const float* pG = rgb + ((size_t)b * 3 + 1) * plane;
    const float* pB = rgb + ((size_t)b * 3 + 2) * plane;

    // ---- cooperative load of 46x46 halo tile (circular wrap), luminance on the fly
    for (int i = tid; i < YDIM * YDIM; i += 128) {
        int rr = i / YDIM;
        int cc = i - rr * YDIM;
        int gr = (row0 + rr - HALO) & MASKW;
        int gc = (col0 + cc - HALO) & MASKW;
        size_t g = (size_t)gr * W_IMG + gc;
        float r = pR[g], gv = pG[g], bv = pB[g];
        int s = rr * Y_STRIDE + cc;
        v4f c4; c4.x = r; c4.y = gv; c4.z = bv; c4.w = 1.0f;
        sC[s] = c4;
        sY[s] = (r + gv + bv) * (1.0f / 3.0f);
    }

    // ---- per-wave 16x16 quadrant of the 32x32 tile
    const int m0 = (wave >> 1) * 16;             // quadrant row origin
    const int n0 = (wave & 1) * 16;              // quadrant col origin
    const int M  = lane & 15;                    // A-matrix row for this lane
    const int k0 = (lane < 16) ? 0 : 2;          // A/B K-pair base per WMMA layout

    // Constant banded A (16x20 split into 5 K=4 chunks): A[M][kk]=1 iff kk in [M, M+4]
    v2f aband[5];
#pragma unroll
    for (int kb = 0; kb < 5; ++kb) {
        int kk0 = 4 * kb + k0;
        aband[kb].x = (kk0     >= M && kk0     <= M + 4) ? 1.0f : 0.0f;
        aband[kb].y = (kk0 + 1 >= M && kk0 + 1 <= M + 4) ? 1.0f : 0.0f;
    }

    // D-matrix layout: lane owns column n0+M, rows mbase..mbase+7
    const int ncol  = n0 + M;
    const int mbase = m0 + ((lane < 16) ? 0 : 8);
    // w = exp(-dist*inv_h) = exp2(C * dist),  C = -inv_h * log2(e)
    const float C_EXP = -(1.0f / (1.0f + 1e-6f)) * 1.4426950408889634f;

    v4f acc[8];
#pragma unroll
    for (int v = 0; v < 8; ++v) acc[v] = (v4f)0.0f;

    __syncthreads();

    for (int s = 0; s < 121; ++s) {
        const int uy = s / 11 - 5;
        const int ux = (s - (s / 11) * 11) - 5;

        // ---- fused pass: rowsum[r][c] = sum_{j=-2..2} (y - y_shift)^2, sliding window.
        // 36 rows x 4 col-groups of 8 = 144 slots over 128 threads.
        for (int slot = tid; slot < ZDIM * 4; slot += 128) {
            int r  = slot >> 2;
            int c0 = (slot & 3) << 3;
            const float* yc = &sY[(r + 5) * Y_STRIDE + (c0 + 5)];
            const float* ys = &sY[(r + 5 + uy) * Y_STRIDE + (c0 + 5 + ux)];
            float z[12];
#pragma unroll
            for (int j = 0; j < 12; ++j) {
                float d = yc[j] - ys[j];
                z[j] = d * d;
            }
            float run = z[0] + z[1] + z[2] + z[3] + z[4];
            float* rs = &sRS[r * RS_STRIDE + c0];
            rs[0] = run;
#pragma unroll
            for (int j = 1; j < 8; ++j) {
                run += z[j + 4] - z[j - 1];
                rs[j] = run;
            }
        }
        __syncthreads();

        // ---- vertical 5-tap as banded matmul: D(16x16) = sum_kb A_kb(16x4) x B_kb(4x16)
        v8f d2 = {0.f, 0.f, 0.f, 0.f, 0.f, 0.f, 0.f, 0.f};
#pragma unroll
        for (int kb = 0; kb < 5; ++kb) {
            int rb = m0 + 4 * kb + k0;           // rowsum array row
            v2f bvec;
            bvec.x = sRS[rb * RS_STRIDE + ncol];
            bvec.y = sRS[(rb + 1) * RS_STRIDE + ncol];
            d2 = __builtin_amdgcn_wmma_f32_16x16x4_f32(
                     false, aband[kb], false, bvec, (short)0, d2, false, false);
        }

        // ---- weights + packed accumulation (lane: fixed column, 8 rows)
        const int nbase = (mbase + uy + HALO) * Y_STRIDE + (ncol + ux + HALO);
#pragma unroll
        for (int v = 0; v < 8; ++v) {
            float w = __builtin_amdgcn_exp2f(
                          C_EXP * __builtin_amdgcn_sqrtf(d2[v]));
            v4f c4 = sC[nbase + v * Y_STRIDE];
            acc[v] += c4 * w;                    // .w accumulates the weight sum
        }
        __syncthreads();
    }

    // ---- normalize and store (acc.w >= 1 because the zero shift contributes w=1)
    float* oR = out + ((size_t)b * 3 + 0) * plane;
    float* oG = out + ((size_t)b * 3 + 1) * plane;
    float* oB = out + ((size_t)b * 3 + 2) * plane;
#pragma unroll
    for (int v = 0; v < 8; ++v) {
        size_t g = (size_t)(row0 + mbase + v) * W_IMG + (col0 + ncol);
        float inv = 1.0f / acc[v].w;
        oR[g] = acc[v].x * inv;
        oG[g] = acc[v].y * inv;
        oB[g] = acc[v].z * inv;
    }
}

extern "C" void kernel_launch(void* const* d_in, const int* in_sizes, int n_in,
                              void* d_out, int out_size, void* d_ws, size_t ws_size,
                              hipStream_t stream) {
    const float* rgb = (const float*)d_in[0];
    float* out = (float*)d_out;
    int batches = in_sizes[0] / (3 * W_IMG * W_IMG);   // 8 for the reference shapes
    dim3 grid(batches * (W_IMG / TILE) * (W_IMG / TILE));  // 8192 workgroups
    nlm_kernel<<<grid, 128, 0, stream>>>(rgb, out);
}